// UpSample_10445360463901
// MI455X (gfx1250) — compile-verified
//
#include <hip/hip_runtime.h>
#include <cmath>

#ifndef M_PI
#define M_PI 3.14159265358979323846
#endif

// ---------------- tuning ----------------
#define TT 1024   // t-samples (input) per block -> 2048 outputs per block
#define NT 256    // threads per block (8 waves on CDNA5 wave32)

typedef __attribute__((ext_vector_type(4))) float v4f;

struct Taps { float a[6]; float b[6]; };  // a: phase out[2t], b: phase out[2t+1]

// CDNA5 async global->LDS path (ASYNCcnt). Guarded so the file always compiles;
// if present we get global_load_async_to_lds_b32 + s_wait_asynccnt in the ISA.
#if defined(__HIP_DEVICE_COMPILE__) && defined(__gfx1250__) && \
    __has_builtin(__builtin_amdgcn_global_load_async_to_lds_b32)
#define USE_ASYNC_LDS 1
#else
#define USE_ASYNC_LDS 0
#endif

#if USE_ASYNC_LDS
typedef __attribute__((address_space(1))) int* gptr_i32;
typedef __attribute__((address_space(3))) int* lptr_i32;
#endif

__global__ __launch_bounds__(NT) void upfir2x_kernel(const float* __restrict__ x,
                                                     float* __restrict__ out,
                                                     int T, Taps tp) {
    __shared__ float s[TT + 8];  // TT + 5 halo, padded

    const int    tid   = threadIdx.x;
    const int    tbase = blockIdx.x * TT;
    const size_t row   = blockIdx.y;

    const float* __restrict__ xr = x   + row * (size_t)T;
    float*       __restrict__ orow = out + row * (size_t)(2 * T);

    // ---- Stage TT+5 inputs covering u = tbase-2 .. tbase+TT+2 into LDS ----
    // Left edge: clamping the ADDRESS to x[0] yields exactly the replicated
    // edge value the reference uses (xx[0] = 2*x[0]; the factor 2 is folded
    // into the taps). Right edge (u >= T) must be ZERO and is patched below.
    for (int i = tid; i < TT + 5; i += NT) {
        int u  = tbase - 2 + i;
        int uc = u < 0 ? 0 : (u >= T ? T - 1 : u);
#if USE_ASYNC_LDS
        __builtin_amdgcn_global_load_async_to_lds_b32(
            (gptr_i32)(xr + uc),
            (lptr_i32)&s[i],
            /*offset=*/0, /*cpol=*/0);
#else
        s[i] = __builtin_nontemporal_load(xr + uc);
#endif
    }
#if USE_ASYNC_LDS
#if __has_builtin(__builtin_amdgcn_s_wait_asynccnt)
    __builtin_amdgcn_s_wait_asynccnt(0);
#else
    asm volatile("s_wait_asynccnt 0" ::: "memory");
#endif
#endif
    __syncthreads();

    // Patch right-edge zeros: reference's edge pad replicates a zero-stuffed
    // odd position there, so samples at u >= T contribute 0.
    {
        const int zfirst = (T - tbase) + 2;  // LDS index of u == T
        if (zfirst < TT + 5) {
            for (int i = zfirst + tid; i < TT + 5; i += NT) s[i] = 0.0f;
        }
    }
    __syncthreads();

    // ---- Compute: each thread does 4 consecutive t, i.e. 8 outputs ----
    const int t0 = tbase + 4 * tid;
    if (t0 < T) {
        const int li = 4 * tid;  // s[li + k] == x[t0 - 2 + k]
        float v[9];
#pragma unroll
        for (int k = 0; k < 9; ++k) v[k] = s[li + k];

        v4f q0, q1;
#pragma unroll
        for (int r = 0; r < 4; ++r) {
            float e0 = tp.a[0] * v[r];
            float e1 = tp.b[0] * v[r];
#pragma unroll
            for (int j = 1; j < 6; ++j) {
                e0 = __builtin_fmaf(tp.a[j], v[r + j], e0);
                e1 = __builtin_fmaf(tp.b[j], v[r + j], e1);
            }
            if (r < 2) { q0[2 * r]       = e0; q0[2 * r + 1]       = e1; }
            else       { q1[2 * (r - 2)] = e0; q1[2 * (r - 2) + 1] = e1; }
        }
        // 2*t0 is a multiple of 8 -> 32-byte aligned; write-once stream => NT stores.
        v4f* op = (v4f*)(orow + 2 * (size_t)t0);
        __builtin_nontemporal_store(q0, op);
        __builtin_nontemporal_store(q1, op + 1);
    }
}

// ---------------- host side: build the Kaiser-sinc taps (double, == numpy) ----
static double bessel_i0(double xv) {
    double s = 1.0, t = 1.0, q = 0.25 * xv * xv;
    for (int k = 1; k < 64; ++k) {
        t *= q / ((double)k * (double)k);
        s += t;
        if (t < s * 1e-17) break;
    }
    return s;
}

static Taps make_taps() {
    const int    K = 12, half = 6;
    const double cutoff = 0.25, half_width = 0.3;
    const double delta_f = 4.0 * half_width;
    const double A = 2.285 * (half - 1) * M_PI * delta_f + 7.95;
    double beta;
    if (A > 50.0)       beta = 0.1102 * (A - 8.7);
    else if (A >= 21.0) beta = 0.5842 * pow(A - 21.0, 0.4) + 0.07886 * (A - 21.0);
    else                beta = 0.0;
    const double alpha = (K - 1) / 2.0;
    const double i0b   = bessel_i0(beta);

    double f[12], sum = 0.0;
    for (int n = 0; n < K; ++n) {
        double c = 1.0 - ((n - alpha) / alpha) * ((n - alpha) / alpha);
        if (c < 0.0) c = 0.0;
        const double w    = bessel_i0(beta * sqrt(c)) / i0b;
        const double tm   = (double)(n - half) + 0.5;      // even-length time grid
        const double xs   = 2.0 * cutoff * tm;
        const double sinc = (xs == 0.0) ? 1.0 : sin(M_PI * xs) / (M_PI * xs);
        f[n] = 2.0 * cutoff * w * sinc;
        sum += f[n];
    }
    Taps tp;
    for (int j = 0; j < 6; ++j) {
        // fold the RATIO=2 zero-stuff gain into the taps (exact power-of-two,
        // bitwise-identical to the reference's 2*x scaling).
        tp.a[j] = (float)(2.0 * f[2 * j + 1] / sum);  // out[2t]   uses odd taps
        tp.b[j] = (float)(2.0 * f[2 * j]     / sum);  // out[2t+1] uses even taps
    }
    return tp;
}

extern "C" void kernel_launch(void* const* d_in, const int* in_sizes, int n_in,
                              void* d_out, int out_size, void* d_ws, size_t ws_size,
                              hipStream_t stream) {
    (void)n_in; (void)out_size; (void)d_ws; (void)ws_size;
    const float* x   = (const float*)d_in[0];
    float*       out = (float*)d_out;

    const int T    = 8192;                // reference: x is [32, 128, 8192]
    const int rows = in_sizes[0] / T;     // B*C = 4096

    const Taps tp = make_taps();
    dim3 grid((T + TT - 1) / TT, rows);
    upfir2x_kernel<<<grid, NT, 0, stream>>>(x, out, T, tp);
}